// ConvBlock_v1_11982958756494
// MI455X (gfx1250) — compile-verified
//
#include <hip/hip_runtime.h>
#include <math.h>

// CDNA5 / gfx1250: wave32, WMMA f32 16x16x4
typedef __attribute__((ext_vector_type(2))) float v2f;
typedef __attribute__((ext_vector_type(8))) float v8f;

#define D 128          // feature dim (D_IN == D_OUT)
#define LN_EPS 1e-5f

// ---------------------------------------------------------------------------
// Zero-init output accumulator and degree array (harness poisons buffers).
__global__ void k_init(float* __restrict__ out, float* __restrict__ deg,
                       int n_out, int n_deg) {
  int i = blockIdx.x * blockDim.x + threadIdx.x;
  if (i < n_out) out[i] = 0.0f;
  if (i < n_deg) deg[i] = 0.0f;
}

// ---------------------------------------------------------------------------
// deg[dst] += 1 for every edge, plus +1 self-loop per node.
__global__ void k_degree(const long long* __restrict__ ei,
                         float* __restrict__ deg, int E, int N) {
  int i = blockIdx.x * blockDim.x + threadIdx.x;
  if (i >= E + N) return;
  int d = (i < E) ? (int)ei[E + i] : (i - E);   // row 1 of edge_index = dst
  atomicAdd(&deg[d], 1.0f);
}

// dinv[v] = deg>0 ? rsqrt(deg) : 0
__global__ void k_dinv(const float* __restrict__ deg, float* __restrict__ dinv,
                       int N) {
  int i = blockIdx.x * blockDim.x + threadIdx.x;
  if (i >= N) return;
  float dg = deg[i];
  dinv[i] = (dg > 0.0f) ? rsqrtf(dg) : 0.0f;
}

// ---------------------------------------------------------------------------
// H = X * W  via V_WMMA_F32_16X16X4_F32.
// One wave32 per 16x16 output tile; block = 8 waves = the 8 column tiles
// of one 16-row strip. K loop: 128 / 4 = 32 WMMA issues per wave.
//
// A (16x4 f32, MxK): lane L -> M = L&15; VGPR0 = K=(L<16?0:2), VGPR1 = K+1.
// B (4x16 f32, KxN): lane L -> N = L&15; VGPR0 = row K=(L<16?0:2), VGPR1 = K+1.
// C/D (16x16 f32):   VGPR r, lanes 0-15 -> M=r, lanes 16-31 -> M=r+8; N=L&15.
__global__ void k_gemm_wmma(const float* __restrict__ X,
                            const float* __restrict__ W,
                            float* __restrict__ H, int N) {
  const int lane  = threadIdx.x & 31;
  const int tileN = threadIdx.x >> 5;      // 0..7  -> column tile (D/16 = 8)
  const int tileM = blockIdx.x;            // row strip of 16 nodes
  const int half  = lane >> 4;             // 0: lanes 0-15, 1: lanes 16-31
  const int m     = lane & 15;

  const int row  = tileM * 16 + m;
  const int rowc = (row < N) ? row : (N - 1);  // clamp loads, keep EXEC all-1s
  const float* xrow = X + (size_t)rowc * D;
  const int col = tileN * 16 + m;

  v8f c = {};
  #pragma unroll 4
  for (int k0 = 0; k0 < D; k0 += 4) {
    const int ka = k0 + 2 * half;
    v2f a, b;
    a.x = xrow[ka];
    a.y = xrow[ka + 1];
    b.x = W[(size_t)ka * D + col];
    b.y = W[(size_t)(ka + 1) * D + col];
    // 8 args: (neg_a, A, neg_b, B, c_mod, C, reuse_a, reuse_b)
    c = __builtin_amdgcn_wmma_f32_16x16x4_f32(false, a, false, b,
                                              (short)0, c, false, false);
  }

  // Store guard is uniform per wave: the whole 16-row strip is in range or
  // we fall back to the predicated tail (never taken when N % 16 == 0).
  float* hbase = H + (size_t)tileM * 16 * D + tileN * 16 + m;
  if (tileM * 16 + 16 <= N) {
    #pragma unroll
    for (int r = 0; r < 8; ++r) {
      hbase[(size_t)(r + 8 * half) * D] = c[r];
    }
  } else {
    #pragma unroll
    for (int r = 0; r < 8; ++r) {
      int orow = tileM * 16 + r + 8 * half;
      if (orow < N) H[(size_t)orow * D + tileN * 16 + m] = c[r];
    }
  }
}

// ---------------------------------------------------------------------------
// Edge aggregation: one wave32 per message (edge or self-loop).
// Each lane moves 4 contiguous floats: coalesced 512B row gather of h[src],
// scaled by dinv[src]*dinv[dst], atomic-added into out[dst].
__global__ void k_scatter(const long long* __restrict__ ei,
                          const float* __restrict__ H,
                          const float* __restrict__ dinv,
                          float* __restrict__ out, int E, int N) {
  long long gid = (long long)blockIdx.x * blockDim.x + threadIdx.x;
  int e    = (int)(gid >> 5);
  int lane = (int)(gid & 31);
  if (e >= E + N) return;

  int s, d;
  if (e < E) { s = (int)ei[e]; d = (int)ei[E + e]; }
  else       { s = e - E;      d = s; }             // self-loop

  float nrm = dinv[s] * dinv[d];
  const float4 hv = *(const float4*)(H + (size_t)s * D + lane * 4);
  float* o = out + (size_t)d * D + lane * 4;
  atomicAdd(o + 0, hv.x * nrm);
  atomicAdd(o + 1, hv.y * nrm);
  atomicAdd(o + 2, hv.z * nrm);
  atomicAdd(o + 3, hv.w * nrm);
}

// ---------------------------------------------------------------------------
// Bias + LayerNorm over feature dim, in place. One wave32 per node row;
// 4 features per lane via b128 load/store; wave32 shuffle-xor reductions.
__global__ void k_layernorm(float* __restrict__ out,
                            const float* __restrict__ b,
                            const float* __restrict__ gamma,
                            const float* __restrict__ beta, int N) {
  int lane = threadIdx.x & 31;
  int row  = blockIdx.x * (blockDim.x >> 5) + (threadIdx.x >> 5);
  if (row >= N) return;

  float* o = out + (size_t)row * D + lane * 4;
  const float4 ov = *(const float4*)o;
  const float4 bv = *(const float4*)(b + lane * 4);
  float v[4] = { ov.x + bv.x, ov.y + bv.y, ov.z + bv.z, ov.w + bv.w };

  float s = v[0] + v[1] + v[2] + v[3];
  #pragma unroll
  for (int off = 16; off >= 1; off >>= 1) s += __shfl_xor(s, off, 32);
  const float mu = s * (1.0f / D);

  float q = 0.0f;
  #pragma unroll
  for (int j = 0; j < 4; ++j) { float t = v[j] - mu; q += t * t; }
  #pragma unroll
  for (int off = 16; off >= 1; off >>= 1) q += __shfl_xor(q, off, 32);
  const float rs = rsqrtf(q * (1.0f / D) + LN_EPS);

  const float4 gv = *(const float4*)(gamma + lane * 4);
  const float4 ev = *(const float4*)(beta + lane * 4);
  float4 r;
  r.x = (v[0] - mu) * rs * gv.x + ev.x;
  r.y = (v[1] - mu) * rs * gv.y + ev.y;
  r.z = (v[2] - mu) * rs * gv.z + ev.z;
  r.w = (v[3] - mu) * rs * gv.w + ev.w;
  *(float4*)o = r;
}

// ---------------------------------------------------------------------------
extern "C" void kernel_launch(void* const* d_in, const int* in_sizes, int n_in,
                              void* d_out, int out_size, void* d_ws, size_t ws_size,
                              hipStream_t stream) {
  const float*     x     = (const float*)d_in[0];      // [N, 128] f32
  const long long* ei    = (const long long*)d_in[1];  // [2, E]  i64
  const float*     W     = (const float*)d_in[2];      // [128, 128] f32
  const float*     b     = (const float*)d_in[3];      // [128] f32
  const float*     gamma = (const float*)d_in[4];      // [128] f32
  const float*     beta  = (const float*)d_in[5];      // [128] f32

  const int N = in_sizes[0] / D;
  const int E = in_sizes[1] / 2;

  float* out = (float*)d_out;
  char*  ws  = (char*)d_ws;
  float* H    = (float*)ws;                                   // N*D floats
  float* deg  = (float*)(ws + (size_t)N * D * sizeof(float)); // N floats
  float* dinv = deg + N;                                      // N floats

  // 1) zero out-accumulator and degree
  {
    int n = N * D;
    k_init<<<(n + 255) / 256, 256, 0, stream>>>(out, deg, n, N);
  }
  // 2) degrees (edges + self-loops)
  {
    int n = E + N;
    k_degree<<<(n + 255) / 256, 256, 0, stream>>>(ei, deg, E, N);
  }
  // 3) dinv
  k_dinv<<<(N + 255) / 256, 256, 0, stream>>>(deg, dinv, N);
  // 4) H = X*W  (WMMA f32): one 16-row strip per block, 8 col-tile waves
  {
    int tilesM = (N + 15) / 16;
    k_gemm_wmma<<<tilesM, 256, 0, stream>>>(x, W, H, N);
  }
  // 5) normalized gather/scatter-add over edges + self-loops
  {
    long long nthreads = (long long)(E + N) * 32;
    int blocks = (int)((nthreads + 255) / 256);
    k_scatter<<<blocks, 256, 0, stream>>>(ei, H, dinv, out, E, N);
  }
  // 6) bias + LayerNorm in place
  k_layernorm<<<(N + 7) / 8, 256, 0, stream>>>(out, b, gamma, beta, N);
}